// GraphSAGEEncoder_30794915512417
// MI455X (gfx1250) — compile-verified
//
#include <hip/hip_runtime.h>

typedef __attribute__((ext_vector_type(2))) float v2f;
typedef __attribute__((ext_vector_type(8))) float v8f;

#define N_NODES 50000
#define N_EDGES 500000
#define IN_DIM 128
#define HID 256

// Guarantee the native no-return global_atomic_add_f32 (L2-resident RMW).
__device__ __forceinline__ void atomic_add_f32(float* p, float v) {
#if defined(__HIP_PLATFORM_AMD__) || defined(__AMDGCN__)
    unsafeAtomicAdd(p, v);               // always lowers to hw f32 fadd atomic
#else
    __hip_atomic_fetch_add(p, v, __ATOMIC_RELAXED, __HIP_MEMORY_SCOPE_AGENT);
#endif
}

// ---------------------------------------------------------------- zero fill
__global__ void zero_f32(float* __restrict__ p, long n) {
    long i = (long)blockIdx.x * blockDim.x + threadIdx.x;
    long stride = (long)gridDim.x * blockDim.x;
    for (; i < n; i += stride) p[i] = 0.0f;
}

// ------------------------------------------------- edge scatter (mean-sum)
// One wave32 per edge. Lane c loads a float4 (global_load_b128) of the source
// row and issues 4 global_atomic_add_f32 into the destination row. The 51MB
// aggregation buffer is L2-resident (192MB L2), so atomics resolve in L2.
__global__ void scatter_add(const float* __restrict__ H,
                            const int* __restrict__ src,
                            const int* __restrict__ dst,
                            float* __restrict__ AGG,
                            float* __restrict__ cnt,
                            int nEdges, int K, int addCount) {
    int gtid = blockIdx.x * blockDim.x + threadIdx.x;
    int e = gtid >> 5;
    int lane = threadIdx.x & 31;
    if (e >= nEdges) return;
    int s = src[e];
    int d = dst[e];
    const float4* hs = (const float4*)(H + (long)s * K);
    float* ad = AGG + (long)d * K;
    int nvec = K >> 2;                       // 32 (K=128) or 64 (K=256)
    for (int c4 = lane; c4 < nvec; c4 += 32) {
        float4 v = hs[c4];
        atomic_add_f32(ad + c4 * 4 + 0, v.x);
        atomic_add_f32(ad + c4 * 4 + 1, v.y);
        atomic_add_f32(ad + c4 * 4 + 2, v.z);
        atomic_add_f32(ad + c4 * 4 + 3, v.w);
    }
    if (addCount && lane == 0) atomic_add_f32(cnt + d, 1.0f);
}

// ------------------------------------------------------------ 1/max(cnt,1)
__global__ void inv_cnt(const float* __restrict__ cnt, float* __restrict__ inv, int n) {
    int i = blockIdx.x * blockDim.x + threadIdx.x;
    if (i < n) inv[i] = 1.0f / fmaxf(cnt[i], 1.0f);
}

// ------------------------------------------ fused dual-GEMM + bias + ReLU
// OUT[m][n] = relu( sum_k AGG[m][k]*inv[m]*Wl[n][k] + sum_k X[m][k]*Wr[n][k] + b[n] )
// One wave per 16x16 output tile, V_WMMA_F32_16X16X4_F32 (exact fp32), K by 4.
// A 16x4 f32 layout (ISA 7.12.2): lane L -> row L&15, K = 2*(L>>4) + {0,1}.
// B 4x16 mirrored; C/D: vgpr v -> row v + 8*(L>>4), col L&15.
__global__ void sage_gemm(const float* __restrict__ X,
                          const float* __restrict__ AGG,
                          const float* __restrict__ inv,
                          const float* __restrict__ Wl,
                          const float* __restrict__ Wr,
                          const float* __restrict__ bias,
                          float* __restrict__ OUT,
                          int K, int tilesN, int tilesM) {
    int wave = (blockIdx.x * blockDim.x + threadIdx.x) >> 5;
    int mt = wave / tilesN;
    int nt = wave % tilesN;
    if (mt >= tilesM) return;                // wave-uniform; EXEC stays all-1s
    int lane = threadIdx.x & 31;
    int half = lane >> 4;                    // 0: K+0/1, 1: K+2/3
    int li   = lane & 15;

    int m = mt * 16 + li;                    // A-row this lane supplies
    int n = nt * 16 + li;                    // B-col this lane supplies
    float scale = inv[m];

    const float* aggRow = AGG + (long)m * K;
    const float* xRow   = X   + (long)m * K;
    const float* wlRow  = Wl  + (long)n * K;
    const float* wrRow  = Wr  + (long)n * K;

    v8f acc = {};
#pragma unroll 4
    for (int k0 = 0; k0 < K; k0 += 4) {
        int kA = k0 + half * 2;
        v2f a0 = *(const v2f*)(aggRow + kA);
        a0.x *= scale; a0.y *= scale;        // fold mean-division into A load
        v2f b0 = *(const v2f*)(wlRow + kA);
        acc = __builtin_amdgcn_wmma_f32_16x16x4_f32(
                  false, a0, false, b0, (short)0, acc, false, false);
        v2f a1 = *(const v2f*)(xRow + kA);
        v2f b1 = *(const v2f*)(wrRow + kA);
        acc = __builtin_amdgcn_wmma_f32_16x16x4_f32(
                  false, a1, false, b1, (short)0, acc, false, false);
    }

    float bv = bias[n];
    int outN = tilesN * 16;
    float* outBase = OUT + (long)(mt * 16 + half * 8) * outN + nt * 16 + li;
#pragma unroll
    for (int v = 0; v < 8; ++v) {
        float r = acc[v] + bv;
        outBase[(long)v * outN] = fmaxf(r, 0.0f);
    }
}

// ---------------------------------------------------------------- launcher
extern "C" void kernel_launch(void* const* d_in, const int* in_sizes, int n_in,
                              void* d_out, int out_size, void* d_ws, size_t ws_size,
                              hipStream_t stream) {
    const float* x   = (const float*)d_in[0];
    const int*   ei  = (const int*)  d_in[1];   // [2, E] flat int32
    const float* W1l = (const float*)d_in[2];
    const float* b1l = (const float*)d_in[3];
    const float* W1r = (const float*)d_in[4];
    const float* W2l = (const float*)d_in[5];
    const float* b2l = (const float*)d_in[6];
    const float* W2r = (const float*)d_in[7];
    float* out = (float*)d_out;

    const int* src = ei;
    const int* dst = ei + N_EDGES;

    float* agg = (float*)d_ws;                          // N*HID floats (51.2 MB)
    float* h1  = agg + (size_t)N_NODES * HID;           // N*HID floats
    float* cnt = h1  + (size_t)N_NODES * HID;           // N floats
    float* inv = cnt + N_NODES;                         // N floats

    const int scatterBlocks = (N_EDGES * 32 + 255) / 256;       // wave per edge
    const int gemmBlocks = (N_NODES / 16) * (HID / 16) / 8;     // 6250, exact

    // ---- layer 1 ----
    zero_f32<<<1024, 256, 0, stream>>>(agg, (long)N_NODES * IN_DIM);
    zero_f32<<<128, 256, 0, stream>>>(cnt, N_NODES);
    scatter_add<<<scatterBlocks, 256, 0, stream>>>(x, src, dst, agg, cnt,
                                                   N_EDGES, IN_DIM, 1);
    inv_cnt<<<(N_NODES + 255) / 256, 256, 0, stream>>>(cnt, inv, N_NODES);
    sage_gemm<<<gemmBlocks, 256, 0, stream>>>(x, agg, inv, W1l, W1r, b1l, h1,
                                              IN_DIM, HID / 16, N_NODES / 16);

    // ---- layer 2 ----
    zero_f32<<<2048, 256, 0, stream>>>(agg, (long)N_NODES * HID);
    scatter_add<<<scatterBlocks, 256, 0, stream>>>(h1, src, dst, agg, nullptr,
                                                   N_EDGES, HID, 0);
    sage_gemm<<<gemmBlocks, 256, 0, stream>>>(h1, agg, inv, W2l, W2r, b2l, out,
                                              HID, HID / 16, N_NODES / 16);
}